// _SASA2d_70892730187898
// MI455X (gfx1250) — compile-verified
//
#include <hip/hip_runtime.h>
#include <math.h>

typedef __attribute__((ext_vector_type(2))) float v2f;
typedef __attribute__((ext_vector_type(8))) float v8f;

#define NB 32
#define NC 64
#define NM 64
#define NH 56
#define NW 56
#define NL 3136   // 56*56

// ---------------------------------------------------------------------------
// Kernel A: per (b, p) compute Gram T = X5 @ Xp^T (64x64, K = 3136 spatial),
// then logits[b,m,p] = wq[m,:] @ T @ wk[m,:]^T + Qsum[b,m]*emb[m,p].
// Grid: 32*9 blocks of 256 threads (8 wave32).
// ---------------------------------------------------------------------------
__global__ __launch_bounds__(256) void sasa_logits_kernel(
    const float* __restrict__ x, const float* __restrict__ wq,
    const float* __restrict__ wk, const float* __restrict__ row_emb,
    const float* __restrict__ col_emb, float* __restrict__ logits)
{
    const int blk = blockIdx.x;
    const int b = blk / 9, p = blk % 9;
    const int dr = p / 3 - 1, dc = p % 3 - 1;   // offset for Xp
    const int t = threadIdx.x;
    const int lane = t & 31, wave = t >> 5;
    const int half = lane >> 4, ln = lane & 15;

    __shared__ float Al[64 * 66];   // x5 chunk, stride 66 (bank-safe, 8B aligned)
    __shared__ float Bl[64 * 66];   // xp chunk
    __shared__ float Wq[64 * 65];   // stride 65: row-per-thread reads conflict-free
    __shared__ float Wk[64 * 65];
    __shared__ float Tl[64 * 64];   // Gram result
    __shared__ float Cs[64];        // per-channel sum of x5

    for (int i = t; i < 64 * 64; i += 256) {
        Wq[(i >> 6) * 65 + (i & 63)] = wq[i];
        Wk[(i >> 6) * 65 + (i & 63)] = wk[i];
    }

    // wave handles 2 of the 16 output 16x16 tiles
    const int tid0 = wave * 2, tid1 = wave * 2 + 1;
    const int tc0 = tid0 >> 2, tn0 = tid0 & 3;
    const int tc1 = tid1 >> 2, tn1 = tid1 & 3;

    v8f acc0 = {}; v8f acc1 = {};
    float csum = 0.f;
    const float* xb = x + (size_t)b * NC * NL;

    for (int ch = 0; ch < 49; ++ch) {          // 49 * 64 = 3136 = L exactly
        const int l0 = ch * 64;
        __syncthreads();
        // Stage 64 channels x 64 spatial positions of both shifted views.
        for (int rep = 0; rep < 16; ++rep) {
            int idx = rep * 256 + t;           // 0..4095
            int c = idx >> 6, kk = idx & 63;
            int l = l0 + kk;
            int i = l / NW, j = l - i * NW;
            // X5: shift (0,+1) with zero pad
            float av = (j + 1 < NW) ? xb[c * NL + i * NW + (j + 1)] : 0.f;
            int gi = i + dr, gj = j + dc;
            float bv = (gi >= 0 && gi < NH && gj >= 0 && gj < NW)
                           ? xb[c * NL + gi * NW + gj] : 0.f;
            Al[c * 66 + kk] = av;
            Bl[c * 66 + kk] = bv;
        }
        __syncthreads();
        if (t < 64) {                          // channel sums of x5 (for Qsum)
            float s = 0.f;
            for (int kk = 0; kk < 64; ++kk) s += Al[t * 66 + kk];
            csum += s;
        }
        // D[m][n] += sum_k A[m][k]*B[k][n]; A-frag K striped over (VGPR, half)
        for (int k0 = 0; k0 < 64; k0 += 4) {
            const int ko = k0 + 2 * half;
            v2f a0 = *(const v2f*)&Al[(tc0 * 16 + ln) * 66 + ko];
            v2f b0 = *(const v2f*)&Bl[(tn0 * 16 + ln) * 66 + ko];
            acc0 = __builtin_amdgcn_wmma_f32_16x16x4_f32(
                false, a0, false, b0, (short)0, acc0, false, false);
            v2f a1 = *(const v2f*)&Al[(tc1 * 16 + ln) * 66 + ko];
            v2f b1 = *(const v2f*)&Bl[(tn1 * 16 + ln) * 66 + ko];
            acc1 = __builtin_amdgcn_wmma_f32_16x16x4_f32(
                false, a1, false, b1, (short)0, acc1, false, false);
        }
    }
    __syncthreads();
    // Spill Gram to LDS per C/D layout: VGPR v, lanes split -> M = v + 8*half
    for (int v = 0; v < 8; ++v) {
        int mlo = v + 8 * half;
        Tl[(tc0 * 16 + mlo) * 64 + tn0 * 16 + ln] = acc0[v];
        Tl[(tc1 * 16 + mlo) * 64 + tn1 * 16 + ln] = acc1[v];
    }
    if (t < 64) Cs[t] = csum;
    __syncthreads();

    if (t < 64) {
        const int m = t;
        float accd = 0.f;
        for (int c = 0; c < 64; ++c) {
            float u = 0.f;
            for (int cp = 0; cp < 64; ++cp) u += Tl[c * 64 + cp] * Wk[m * 65 + cp];
            accd += Wq[m * 65 + c] * u;
        }
        float qs = 0.f;
        for (int c = 0; c < 64; ++c) qs += Wq[m * 65 + c] * Cs[c];
        float embv = (m < 32) ? row_emb[(p / 3) * 32 + m]
                              : col_emb[(p % 3) * 32 + (m - 32)];
        logits[(b * 64 + m) * 9 + p] = accd + qs * embv;
    }
}

// ---------------------------------------------------------------------------
// Kernel B: softmax over the 9 offsets per (b, m). 2048 independent rows.
// ---------------------------------------------------------------------------
__global__ void sasa_softmax_kernel(const float* __restrict__ logits,
                                    float* __restrict__ att)
{
    int n = blockIdx.x * blockDim.x + threadIdx.x;
    if (n >= NB * NM) return;
    float v[9];
    float mx = -3.0e38f;
    for (int p = 0; p < 9; ++p) { v[p] = logits[n * 9 + p]; mx = fmaxf(mx, v[p]); }
    float s = 0.f;
    for (int p = 0; p < 9; ++p) { v[p] = expf(v[p] - mx); s += v[p]; }
    float inv = 1.f / s;
    for (int p = 0; p < 9; ++p) att[n * 9 + p] = v[p] * inv;
}

// ---------------------------------------------------------------------------
// Kernel C: out[b] = Weff[b] @ unfold(x[b]); Weff[m][(p,c)] = att[b,m,p]*wv[m,c]
// GEMM M=64, N=3136 (tiled by 128 pixels), K=576 (chunked by offset p).
// Grid: 32 * 25 blocks of 256 threads; each wave owns 4 of 32 16x16 tiles.
// ---------------------------------------------------------------------------
__global__ __launch_bounds__(256) void sasa_out_kernel(
    const float* __restrict__ x, const float* __restrict__ wv,
    const float* __restrict__ att, float* __restrict__ out)
{
    const int nT = 25;                       // ceil(3136/128)
    const int blk = blockIdx.x;
    const int b = blk / nT, tt = blk % nT;
    const int l0 = tt * 128;
    const int t = threadIdx.x;
    const int lane = t & 31, wave = t >> 5;
    const int half = lane >> 4, ln = lane & 15;

    __shared__ float Wv[64 * 66];            // stride 66 -> conflict-free b64 reads
    __shared__ float Bl[64 * 128];           // xp chunk: [c][pixel]
    __shared__ float At[9 * 64];             // att[p][m]

    for (int i = t; i < 64 * 64; i += 256)
        Wv[(i >> 6) * 66 + (i & 63)] = wv[i];
    for (int i = t; i < 576; i += 256) {
        int p = i >> 6, m = i & 63;
        At[p * 64 + m] = att[(b * 64 + m) * 9 + p];
    }

    v8f acc[4] = {v8f{}, v8f{}, v8f{}, v8f{}};
    const float* xb = x + (size_t)b * NC * NL;

    for (int p = 0; p < 9; ++p) {
        const int dr = p / 3 - 1, dc = p % 3 - 1;
        __syncthreads();
        for (int rep = 0; rep < 32; ++rep) {
            int idx = rep * 256 + t;          // 0..8191
            int c = idx >> 7, px = idx & 127;
            int l = l0 + px;
            int i = l / NW, j = l - i * NW;
            int gi = i + dr, gj = j + dc;
            float v = (l < NL && gi >= 0 && gi < NH && gj >= 0 && gj < NW)
                          ? xb[c * NL + gi * NW + gj] : 0.f;
            Bl[c * 128 + px] = v;
        }
        __syncthreads();
        for (int i2 = 0; i2 < 4; ++i2) {
            const int tid = wave * 4 + i2;
            const int tc = tid >> 3, tn = tid & 7;
            const int m = tc * 16 + ln;
            const float attv = At[p * 64 + m];
            for (int k0 = 0; k0 < 64; k0 += 4) {
                const int ko = k0 + 2 * half;
                v2f wv2 = *(const v2f*)&Wv[m * 66 + ko];
                v2f a; a.x = attv * wv2.x; a.y = attv * wv2.y;
                v2f bf;
                bf.x = Bl[ko * 128 + tn * 16 + ln];
                bf.y = Bl[(ko + 1) * 128 + tn * 16 + ln];
                acc[i2] = __builtin_amdgcn_wmma_f32_16x16x4_f32(
                    false, a, false, bf, (short)0, acc[i2], false, false);
            }
        }
    }
    // Store per C/D layout: VGPR v -> row tc*16 + v + 8*half, col = lane
    for (int i2 = 0; i2 < 4; ++i2) {
        const int tid = wave * 4 + i2;
        const int tc = tid >> 3, tn = tid & 7;
        for (int v = 0; v < 8; ++v) {
            int m = tc * 16 + v + 8 * half;
            int l = l0 + tn * 16 + ln;
            if (l < NL) out[((size_t)(b * 64 + m)) * NL + l] = acc[i2][v];
        }
    }
}

extern "C" void kernel_launch(void* const* d_in, const int* in_sizes, int n_in,
                              void* d_out, int out_size, void* d_ws, size_t ws_size,
                              hipStream_t stream) {
    const float* x       = (const float*)d_in[0];
    const float* wq      = (const float*)d_in[1];
    const float* wk      = (const float*)d_in[2];
    const float* wv      = (const float*)d_in[3];
    const float* row_emb = (const float*)d_in[4];
    const float* col_emb = (const float*)d_in[5];
    float* out = (float*)d_out;

    float* logits = (float*)d_ws;                // 32*64*9 floats
    float* att    = logits + NB * NM * 9;        // 32*64*9 floats

    sasa_logits_kernel<<<NB * 9, 256, 0, stream>>>(x, wq, wk, row_emb, col_emb, logits);
    sasa_softmax_kernel<<<(NB * NM + 255) / 256, 256, 0, stream>>>(logits, att);
    sasa_out_kernel<<<NB * 25, 256, 0, stream>>>(x, wv, att, out);
}